// Quantize_22754736734271
// MI455X (gfx1250) — compile-verified
//
#include <hip/hip_runtime.h>

// CDNA5 / gfx1250 VQ-VAE quantize.
// GEMM via V_WMMA_F32_16X16X4_F32; argmin fused into the WMMA epilogue.
// score = ||e||^2 - 2 x.e  (||x||^2 dropped: constant per row for argmin).
// A is pre-scaled by -2 (exact), C seeded with ||e||^2, so the WMMA output
// is the score directly. Each M-block's K=64 reduction is split into two
// independent 8-deep WMMA chains for ILP (4 chains per wave total).

typedef __attribute__((ext_vector_type(2))) float v2f;
typedef __attribute__((ext_vector_type(8))) float v8f;

#define DIM    64
#define NE     1024
#define NROWS  (32 * 64 * 64)           // 131072 rows
#define QN     (NROWS * DIM)            // 8388608 quantize elements
#define DIFF_OFF  QN                    // scalar diff
#define IND_OFF   (QN + 1)              // 131072 indices (as float)
#define NUSED_OFF (IND_OFF + NROWS)     // scalar -1

// ---------------------------------------------------------------------------
// Kernel 1: code-vector squared norms ||e_j||^2, j = 0..1023
// ---------------------------------------------------------------------------
__global__ __launch_bounds__(1024) void vq_enorm(const float* __restrict__ embed,
                                                 float* __restrict__ enorm) {
    const int j = threadIdx.x;  // 1024 threads, 1 block
    float s = 0.0f;
#pragma unroll
    for (int d = 0; d < DIM; ++d) {
        float v = embed[d * NE + j];
        s = fmaf(v, v, s);
    }
    enorm[j] = s;
}

// ---------------------------------------------------------------------------
// Kernel 2: main GEMM + argmin + gather + local diff partials
// Block = 256 threads = 8 waves; wave owns 32 rows (two 16-row M blocks).
// Grid = NROWS / 256 = 512 blocks.
// ---------------------------------------------------------------------------
__global__ __launch_bounds__(256) void vq_main(const float* __restrict__ input,
                                               const float* __restrict__ embed,
                                               const float* __restrict__ enorm,
                                               float* __restrict__ out,
                                               float* __restrict__ partials) {
    const int lane = threadIdx.x & 31;
    const int wave = threadIdx.x >> 5;
    const int hi   = lane >> 4;   // 0: lanes 0-15, 1: lanes 16-31
    const int lo   = lane & 15;
    const int base = blockIdx.x * 256 + wave * 32;   // first of 32 rows

    // ---- Preload A fragments, pre-scaled by -2 (exact scaling) -------------
    // f32 16x16x4 A layout: lane = M (mod 16); VGPR pair = K = k0 + 2*hi + {0,1}.
    float a0[32], a1[32];
    {
        const float* r0 = input + (size_t)(base + lo) * DIM + 2 * hi;
        const float* r1 = input + (size_t)(base + 16 + lo) * DIM + 2 * hi;
#pragma unroll
        for (int s = 0; s < 16; ++s) {
            float2 t0 = *(const float2*)(r0 + 4 * s);
            float2 t1 = *(const float2*)(r1 + 4 * s);
            a0[2 * s]     = -2.0f * t0.x;
            a0[2 * s + 1] = -2.0f * t0.y;
            a1[2 * s]     = -2.0f * t1.x;
            a1[2 * s + 1] = -2.0f * t1.y;
        }
    }

    // Running argmin state: C VGPR r holds row r (lanes 0-15) / row 8+r (16-31),
    // column N = j0 + lo. 8 slots per M block.
    float minv0[8], minv1[8];
    int   mini0[8], mini1[8];
#pragma unroll
    for (int r = 0; r < 8; ++r) {
        minv0[r] = 3.4028235e38f; minv1[r] = 3.4028235e38f;
        mini0[r] = 0;             mini1[r] = 0;
    }

    // ---- Sweep all 1024 codes in 16-wide N tiles ---------------------------
    for (int j0 = 0; j0 < NE; j0 += 16) {
        const int jc = j0 + lo;          // this lane's code column
        const float en = enorm[jc];      // same N for all 8 C slots of a lane
        // Four independent accumulator chains (2 per M block) for ILP:
        // chain A covers K=0..31 (seeded with ||e||^2), chain B covers K=32..63.
        v8f c0a = {en, en, en, en, en, en, en, en};
        v8f c1a = c0a;
        v8f c0b = {0.f, 0.f, 0.f, 0.f, 0.f, 0.f, 0.f, 0.f};
        v8f c1b = c0b;

#pragma unroll
        for (int s = 0; s < 8; ++s) {    // chains A: K = 4s..4s+3, B: K = 32+4s..
            // B 4x16 layout: lane = N (mod 16); VGPR pair = K = k0 + 2*hi + {0,1}.
            v2f bfa, bfb;
            bfa.x = embed[(size_t)(4 * s + 2 * hi) * NE + jc];
            bfa.y = embed[(size_t)(4 * s + 2 * hi + 1) * NE + jc];
            bfb.x = embed[(size_t)(32 + 4 * s + 2 * hi) * NE + jc];
            bfb.y = embed[(size_t)(32 + 4 * s + 2 * hi + 1) * NE + jc];
            v2f a0a; a0a.x = a0[2 * s];      a0a.y = a0[2 * s + 1];
            v2f a0b; a0b.x = a0[16 + 2 * s]; a0b.y = a0[16 + 2 * s + 1];
            v2f a1a; a1a.x = a1[2 * s];      a1a.y = a1[2 * s + 1];
            v2f a1b; a1b.x = a1[16 + 2 * s]; a1b.y = a1[16 + 2 * s + 1];
            c0a = __builtin_amdgcn_wmma_f32_16x16x4_f32(
                      false, a0a, false, bfa, (short)0, c0a, false, false);
            c1a = __builtin_amdgcn_wmma_f32_16x16x4_f32(
                      false, a1a, false, bfa, (short)0, c1a, false, false);
            c0b = __builtin_amdgcn_wmma_f32_16x16x4_f32(
                      false, a0b, false, bfb, (short)0, c0b, false, false);
            c1b = __builtin_amdgcn_wmma_f32_16x16x4_f32(
                      false, a1b, false, bfb, (short)0, c1b, false, false);
        }

#pragma unroll
        for (int r = 0; r < 8; ++r) {
            float s0 = c0a[r] + c0b[r];
            if (s0 < minv0[r]) { minv0[r] = s0; mini0[r] = jc; }
            float s1 = c1a[r] + c1b[r];
            if (s1 < minv1[r]) { minv1[r] = s1; mini1[r] = jc; }
        }
    }

    // ---- Cross-lane argmin within each 16-lane half (xor 1,2,4,8) ----------
    // Tie-break to the lower code index, matching jnp.argmin semantics.
#pragma unroll
    for (int r = 0; r < 8; ++r) {
#pragma unroll
        for (int m = 1; m <= 8; m <<= 1) {
            float ov = __shfl_xor(minv0[r], m, 32);
            int   oi = __shfl_xor(mini0[r], m, 32);
            if (ov < minv0[r] || (ov == minv0[r] && oi < mini0[r])) {
                minv0[r] = ov; mini0[r] = oi;
            }
            ov = __shfl_xor(minv1[r], m, 32);
            oi = __shfl_xor(mini1[r], m, 32);
            if (ov < minv1[r] || (ov == minv1[r] && oi < mini1[r])) {
                minv1[r] = ov; mini1[r] = oi;
            }
        }
    }

    // ---- Write indices (as float, per output dtype) ------------------------
    // Lanes 0-15 hold rows {r}, lanes 16-31 hold rows {8+r} of each M block.
    if (lo == 0) {
#pragma unroll
        for (int r = 0; r < 8; ++r) {
            out[IND_OFF + base + r + 8 * hi]      = (float)mini0[r];
            out[IND_OFF + base + 16 + r + 8 * hi] = (float)mini1[r];
        }
    }

    // ---- Gather winning codes, write straight-through output, diff ---------
    // Reference returns x + (q - x) in f32 (not bitwise q); we reproduce it
    // exactly by reusing the d = q - x subtraction from the diff term.
    float dacc = 0.0f;
#pragma unroll
    for (int m = 0; m < 16; ++m) {
        const int src = (m < 8) ? 0 : 16;
        const int idx0 = __shfl(mini0[m & 7], src, 32);
        const int idx1 = __shfl(mini1[m & 7], src, 32);
        {
            const int row = base + m;
            float q0 = embed[(size_t)lane * NE + idx0];
            float q1 = embed[(size_t)(lane + 32) * NE + idx0];
            float x0 = input[(size_t)row * DIM + lane];
            float x1 = input[(size_t)row * DIM + lane + 32];
            float d0 = q0 - x0, d1 = q1 - x1;
            out[(size_t)row * DIM + lane]      = x0 + d0;
            out[(size_t)row * DIM + lane + 32] = x1 + d1;
            dacc = fmaf(d0, d0, dacc);
            dacc = fmaf(d1, d1, dacc);
        }
        {
            const int row = base + 16 + m;
            float q0 = embed[(size_t)lane * NE + idx1];
            float q1 = embed[(size_t)(lane + 32) * NE + idx1];
            float x0 = input[(size_t)row * DIM + lane];
            float x1 = input[(size_t)row * DIM + lane + 32];
            float d0 = q0 - x0, d1 = q1 - x1;
            out[(size_t)row * DIM + lane]      = x0 + d0;
            out[(size_t)row * DIM + lane + 32] = x1 + d1;
            dacc = fmaf(d0, d0, dacc);
            dacc = fmaf(d1, d1, dacc);
        }
    }

    // ---- Deterministic block partial sum -----------------------------------
#pragma unroll
    for (int m = 1; m < 32; m <<= 1) dacc += __shfl_xor(dacc, m, 32);

    __shared__ float sacc[8];
    if (lane == 0) sacc[wave] = dacc;
    __syncthreads();
    if (threadIdx.x == 0) {
        float t = 0.0f;
#pragma unroll
        for (int w = 0; w < 8; ++w) t += sacc[w];
        partials[blockIdx.x] = t;
    }
}

// ---------------------------------------------------------------------------
// Kernel 3: deterministic reduction of 512 block partials -> diff, plus -1.
// ---------------------------------------------------------------------------
__global__ __launch_bounds__(512) void vq_finalize(const float* __restrict__ partials,
                                                   float* __restrict__ out) {
    __shared__ float s[512];
    const int t = threadIdx.x;
    s[t] = partials[t];
    __syncthreads();
#pragma unroll
    for (int off = 256; off > 0; off >>= 1) {
        if (t < off) s[t] += s[t + off];
        __syncthreads();
    }
    if (t == 0) {
        out[DIFF_OFF]  = s[0] / (float)QN;
        out[NUSED_OFF] = -1.0f;
    }
}

// ---------------------------------------------------------------------------
extern "C" void kernel_launch(void* const* d_in, const int* in_sizes, int n_in,
                              void* d_out, int out_size, void* d_ws, size_t ws_size,
                              hipStream_t stream) {
    (void)in_sizes; (void)n_in; (void)out_size; (void)ws_size;
    const float* input = (const float*)d_in[0];   // [131072, 64]
    const float* embed = (const float*)d_in[1];   // [64, 1024]
    float* out = (float*)d_out;

    float* enorm    = (float*)d_ws;        // 1024 floats
    float* partials = enorm + NE;          // 512 floats

    vq_enorm   <<<1,   1024, 0, stream>>>(embed, enorm);
    vq_main    <<<512, 256,  0, stream>>>(input, embed, enorm, out, partials);
    vq_finalize<<<1,   512,  0, stream>>>(partials, out);
}